// NeuralSheafDiffusion_34248069219259
// MI455X (gfx1250) — compile-verified
//
#include <hip/hip_runtime.h>
#include <hip/hip_bf16.h>
#include <stdint.h>

typedef float v2f __attribute__((ext_vector_type(2)));
typedef float v8f __attribute__((ext_vector_type(8)));

__device__ __forceinline__ v8f wmma_f32_16x16x4(v2f a, v2f b, v8f c) {
  // D = A(16x4, f32) * B(4x16, f32) + C(16x16, f32)
  return __builtin_amdgcn_wmma_f32_16x16x4_f32(false, a, false, b, (short)0, c,
                                               false, false);
}

__device__ __forceinline__ void lds_fence() {
  asm volatile("s_wait_dscnt 0" ::: "memory");
}

__device__ __forceinline__ void atomic_add_agent(float* p, float v) {
  // device-scope relaxed fp add -> native global_atomic_add_f32 on CDNA5
  __hip_atomic_fetch_add(p, v, __ATOMIC_RELAXED, __HIP_MEMORY_SCOPE_AGENT);
}

// ---------------- kernel 1: init agg=0, deg=1 (self loop) ----------------
__global__ void k_init(float* __restrict__ agg, float* __restrict__ deg, int n) {
  int idx = blockIdx.x * blockDim.x + threadIdx.x;
  if (idx < n * 64) agg[idx] = 0.0f;
  if (idx < n) deg[idx] = 1.0f;
}

// ---------------- kernel 2: degree over source nodes ----------------
__global__ void k_degree(const long long* __restrict__ ei,
                         float* __restrict__ deg, int E) {
  int e = blockIdx.x * blockDim.x + threadIdx.x;
  if (e < E) atomic_add_agent(&deg[(int)ei[e]], 1.0f);
}

// ---------------- kernel 3: dis = deg^-1/2 (deg>=1 always) ----------------
__global__ void k_dis(float* __restrict__ deg, int n) {
  int i = blockIdx.x * blockDim.x + threadIdx.x;
  if (i < n) deg[i] = rsqrtf(deg[i]);
}

// ---------------- kernel 4: xf = (W1^T xr) W2 per node ----------------
__global__ void __launch_bounds__(256)
k_node(const float* __restrict__ x, const float* __restrict__ W1,
       const float* __restrict__ W2, float* __restrict__ xf, int n) {
  __shared__ float xs[4 * 64];
  __shared__ float w1s[16];
  __shared__ float w2s[256];
  const int t = threadIdx.x;
  if (t < 16) w1s[t] = W1[t];
  w2s[t] = W2[t];
  const int nBase = blockIdx.x * 4;
  const int gid = nBase * 64 + t;
  xs[t] = (gid < n * 64) ? x[gid] : 0.0f;
  __syncthreads();
  const int nl = t >> 6;
  const int k = t & 63;
  const int d = k >> 4;
  const int g = k & 15;
  const int node = nBase + nl;
  if (node < n) {
    const float* xr = &xs[nl * 64];
    float acc = 0.0f;
    for (int f = 0; f < 16; ++f) {
      float tdf = 0.0f;
      for (int dd = 0; dd < 4; ++dd) tdf += xr[dd * 16 + f] * w1s[dd * 4 + d];
      acc += tdf * w2s[f * 16 + g];
    }
    xf[(long long)node * 64 + d * 16 + g] = acc;
  }
}

// ---------------- kernel 5: per-edge message kernel (WMMA) ----------------
// 128 threads = 4 wave32; persistent: each wave grid-strides over 16-edge tiles.
// B fragments (mlp1_w) are preloaded into 64 VGPRs once per wave.
__global__ void __launch_bounds__(128)
k_edge(const float* __restrict__ xf, const long long* __restrict__ ei,
       const float* __restrict__ dv, const float* __restrict__ mlp1_w,
       const float* __restrict__ mlp1_b, const float* __restrict__ mlp2_w,
       const float* __restrict__ mlp2_b, float* __restrict__ agg,
       int E, int Et, int numTiles) {
  __shared__ float w1s[16 * 128];
  __shared__ float b1s[16];
  __shared__ float w2s[6 * 16];
  __shared__ float b2s[6];
  __shared__ alignas(16) float cat_s[4][16][132];  // [wave][edge][xi(64)|xj(64)], padded
  __shared__ float Hs[4][2][16][17];               // relu hidden, [wave][ve/ue][edge][h]
  __shared__ float Fs[4][2][16][16];               // Q matrices, [..][edge][k*4+a]
  __shared__ float Gs[4][16][16];                  // Fve^T Fue,  [wave][edge][a*4+b]
  __shared__ int   colI[4][16];
  __shared__ int   rowI[4][16];
  __shared__ float scl[4][16];

  const int t = threadIdx.x;
  for (int i = t; i < 16 * 128; i += 128) w1s[i] = mlp1_w[i];
  if (t < 16) b1s[t] = mlp1_b[t];
  if (t < 96) w2s[t] = mlp2_w[t];
  if (t < 6)  b2s[t] = mlp2_b[t];
  __syncthreads();

  const int w = t >> 5;        // wave in block (wave32!)
  const int lane = t & 31;
  const int m = lane & 15;                 // A row (edge) / B col (out channel)
  const int khalf = (lane >> 4) << 1;      // lanes 16-31 hold K+2,K+3

  // ---- hoist B fragments (constant across tiles): 32 chunks x v2f ----
  v2f Bf[32];
#pragma unroll
  for (int c = 0; c < 32; ++c) {
    const int k0 = c * 4 + khalf;
    Bf[c].x = w1s[m * 128 + k0];
    Bf[c].y = w1s[m * 128 + k0 + 1];
  }

  for (int tile = blockIdx.x * 4 + w; tile < numTiles;
       tile += (int)gridDim.x * 4) {
    const long long eBase = (long long)tile * 16;

    // ---- edge meta: row (source j), col (target i), scale ----
    if (lane < 16) {
      long long e = eBase + lane;
      int r = 0, c = 0;
      float s = 0.0f;
      if (e < (long long)E) {
        r = (int)ei[e];                 // edge_index[0][e]
        c = (int)ei[(long long)E + e];  // edge_index[1][e]
        s = dv[c] * dv[r];
      } else if (e < (long long)Et) {
        r = c = (int)(e - E);           // self loop
        float d0 = dv[r];
        s = d0 * d0;
      }
      rowI[w][lane] = r;
      colI[w][lane] = c;
      scl[w][lane]  = s;
    }
    lds_fence();

    // ---- gather: cat[e][0:64]=xf[col] (xi), cat[e][64:128]=xf[row] (xj) ----
    {
      const int q = lane & 15;                 // float4 slot within a 64-f row
      const int half = (lane < 16) ? 0 : 64;
      for (int e = 0; e < 16; ++e) {
        const int node = (lane < 16) ? colI[w][e] : rowI[w][e];
        const float4 v = ((const float4*)(xf + (long long)node * 64))[q];
        *(float4*)(&cat_s[w][e][half + q * 4]) = v;
      }
    }
    lds_fence();

    // ---- layer-1 GEMM: [16 edges x 128] x [128 x 16], 32 K=4 WMMA chunks.
    // cat_ue(e,k) = cat_ve(e, k XOR 64): one LDS buffer serves both MLPs.
    v8f cve = {0.f, 0.f, 0.f, 0.f, 0.f, 0.f, 0.f, 0.f};
    v8f cue = {0.f, 0.f, 0.f, 0.f, 0.f, 0.f, 0.f, 0.f};
#pragma unroll
    for (int c = 0; c < 32; ++c) {
      const int k0 = c * 4 + khalf;
      v2f a;  a.x  = cat_s[w][m][k0];        a.y  = cat_s[w][m][k0 + 1];
      cve = wmma_f32_16x16x4(a, Bf[c], cve);
      v2f a2; a2.x = cat_s[w][m][k0 ^ 64];   a2.y = cat_s[w][m][(k0 + 1) ^ 64];
      cue = wmma_f32_16x16x4(a2, Bf[c], cue);
    }

    // ---- bias + relu -> LDS (C layout: vgpr r = edge r (lo) / r+8 (hi)) ----
    {
      const int h = lane & 15;
      const int ebase = (lane >> 4) << 3;
#pragma unroll
      for (int r = 0; r < 8; ++r) {
        const int e = ebase + r;
        float hv = cve[r] + b1s[h];
        float hu = cue[r] + b1s[h];
        Hs[w][0][e][h] = hv > 0.f ? hv : 0.f;
        Hs[w][1][e][h] = hu > 0.f ? hu : 0.f;
      }
    }
    lds_fence();

    // ---- layer-2 (16->6) + QR per (edge, side); one lane each ----
    {
      const int side = lane >> 4;
      const int e = lane & 15;
      const float* Hrow = Hs[w][side][e];
      float p[6];
#pragma unroll
      for (int pi = 0; pi < 6; ++pi) {
        float acc = b2s[pi];
        for (int h = 0; h < 16; ++h) acc += w2s[pi * 16 + h] * Hrow[h];
        p[pi] = acc;
      }
      // A = I + strictly-lower(params), tril order (1,0)(2,0)(2,1)(3,0)(3,1)(3,2)
      float A[4][4] = {{1.f, 0.f, 0.f, 0.f},
                       {p[0], 1.f, 0.f, 0.f},
                       {p[1], p[2], 1.f, 0.f},
                       {p[3], p[4], p[5], 1.f}};
      float Q[4][4];
      for (int j = 0; j < 4; ++j) {     // modified Gram-Schmidt; diag(R) > 0
        float v[4] = {A[0][j], A[1][j], A[2][j], A[3][j]};
        for (int i = 0; i < j; ++i) {
          float dot = 0.f;
          for (int k2 = 0; k2 < 4; ++k2) dot += Q[k2][i] * v[k2];
          for (int k2 = 0; k2 < 4; ++k2) v[k2] -= dot * Q[k2][i];
        }
        float nrm = rsqrtf(v[0]*v[0] + v[1]*v[1] + v[2]*v[2] + v[3]*v[3]);
        for (int k2 = 0; k2 < 4; ++k2) Q[k2][j] = v[k2] * nrm;
      }
      for (int k2 = 0; k2 < 4; ++k2)
        for (int a = 0; a < 4; ++a)
          Fs[w][side][e][k2 * 4 + a] = Q[k2][a];
    }
    lds_fence();

    // ---- G = Fve^T Fue per edge ----
    if (lane < 16) {
      const int e = lane;
      for (int a = 0; a < 4; ++a)
        for (int b = 0; b < 4; ++b) {
          float acc = 0.f;
          for (int k2 = 0; k2 < 4; ++k2)
            acc += Fs[w][0][e][k2 * 4 + a] * Fs[w][1][e][k2 * 4 + b];
          Gs[w][e][a * 4 + b] = acc;
        }
    }
    lds_fence();

    // ---- msg[a][h] = s * sum_b G[a][b]*xj[b*16+h]; scatter-add to agg[col] ----
    for (int e = 0; e < 16; ++e) {
      const float sc = scl[w][e];
      const long long node = colI[w][e];
#pragma unroll
      for (int it = 0; it < 2; ++it) {
        const int k = it * 32 + lane;
        const int a = k >> 4, h = k & 15;
        float acc = 0.f;
        for (int b = 0; b < 4; ++b)
          acc += Gs[w][e][a * 4 + b] * cat_s[w][e][64 + b * 16 + h];
        atomic_add_agent(&agg[node * 64 + k], sc * acc);
      }
    }
    lds_fence();  // protect cat_s/meta reuse on next grid-stride iteration
  }
}

// ---------------- kernel 6: out = coeff*x - elu(agg) ----------------
__global__ void k_out(const float* __restrict__ x, const float* __restrict__ agg,
                      const float* __restrict__ eps, float* __restrict__ out,
                      int n) {
  int idx = blockIdx.x * blockDim.x + threadIdx.x;
  if (idx >= n * 64) return;
  const int d = (idx >> 4) & 3;
  const float c = 1.0f + tanhf(eps[d]);
  const float a = agg[idx];
  const float lap = a > 0.f ? a : (expf(a) - 1.0f);
  out[idx] = c * x[idx] - lap;
}

extern "C" void kernel_launch(void* const* d_in, const int* in_sizes, int n_in,
                              void* d_out, int out_size, void* d_ws,
                              size_t ws_size, hipStream_t stream) {
  const float*     x   = (const float*)d_in[0];
  const long long* ei  = (const long long*)d_in[1];   // int64 [2, E]
  const float*     W1  = (const float*)d_in[2];
  const float*     W2  = (const float*)d_in[3];
  const float*     eps = (const float*)d_in[4];
  const float*     m1w = (const float*)d_in[5];
  const float*     m1b = (const float*)d_in[6];
  const float*     m2w = (const float*)d_in[7];
  const float*     m2b = (const float*)d_in[8];
  float* out = (float*)d_out;

  const int N  = in_sizes[0] / 64;
  const int E  = in_sizes[1] / 2;
  const int Et = E + N;

  // workspace: [deg/dis N][xf N*64][agg N*64]  ~= 26 MB
  float* deg = (float*)d_ws;
  float* xf  = deg + N;
  float* agg = xf + (size_t)N * 64;

  k_init<<<(N * 64 + 255) / 256, 256, 0, stream>>>(agg, deg, N);
  k_degree<<<(E + 255) / 256, 256, 0, stream>>>(ei, deg, E);
  k_dis<<<(N + 255) / 256, 256, 0, stream>>>(deg, N);
  k_node<<<(N + 3) / 4, 256, 0, stream>>>(x, W1, W2, xf, N);

  const int numTiles = (Et + 15) / 16;   // ~53125 tiles of 16 edges
  int blocks = (numTiles + 3) / 4;
  if (blocks > 4096) blocks = 4096;      // persistent waves, ~3.2 tiles each
  k_edge<<<blocks, 128, 0, stream>>>(xf, ei, deg, m1w, m1b, m2w, m2b, agg, E,
                                     Et, numTiles);

  k_out<<<(N * 64 + 255) / 256, 256, 0, stream>>>(x, agg, eps, out, N);
}